// CLDiceLoss_50955491999976
// MI455X (gfx1250) — compile-verified
//
#include <hip/hip_runtime.h>
#include <hip/hip_bf16.h>
#include <stdint.h>

// ---------------------------------------------------------------------------
// clDice loss, MI455X (gfx1250).
// Memory-bound morphology. Hot iterate kernel: one Tensor Data Mover
// descriptor (tensor_load_to_lds + s_wait_tensorcnt) stages a 40x14x14 f32
// tile to LDS; init kernel uses GLOBAL_LOAD_ASYNC_TO_LDS_B128
// (+ s_wait_asynccnt). Separable 3x3x3 min/max morphology entirely in LDS.
// Output tile 32x8x8 -> global read amplification 3.83x (vs 6.1x for 8^3).
// Deterministic two-stage reduction.
// Workspace: 12*N + 4*NRBLK floats (~340 MB), N = 192^3.
// ---------------------------------------------------------------------------

#define VD    192
#define NVOX  (192*192*192)
#define TX    32              // output tile x (contiguous axis)
#define TZY   8               // output tile z,y
#define PIT   40              // LDS x pitch == TDM tile_dim0 (halo-4 .. halo+35)
#define SBLK  512             // stencil block: 16 wave32
#define NTHR  256             // elementwise/reduce block
#define NRBLK 4096
#define PADPOS ( 3.0e38f)
#define PADNEG (-3.0e38f)

typedef __attribute__((ext_vector_type(4))) unsigned int u32x4;
typedef __attribute__((ext_vector_type(8))) unsigned int u32x8;

__device__ __forceinline__ float f3min(float a,float b,float c){return fminf(a,fminf(b,c));}
__device__ __forceinline__ float f3max(float a,float b,float c){return fmaxf(a,fmaxf(b,c));}

// Column c maps to global x = x0 - 4 + c. Each x-reduction narrows the valid
// column window by one on each side: IN [1,38] -> E1 [2,37] -> E2 [3,36] -> O [4,35].

template<bool MN>
__device__ __forceinline__ void red_z(const float* __restrict__ s, float* __restrict__ d,
                                      int nz,int ny,int xlo,int xhi,int tid){
  const int nx = xhi-xlo+1, tot = nz*ny*nx, st = ny*PIT;
  for (int i=tid;i<tot;i+=SBLK){
    int x=xlo+i%nx; int r=i/nx; int y=r%ny; int z=r/ny;
    int o=(z*ny+y)*PIT+x;
    float a=s[o],b=s[o+st],c=s[o+2*st];
    d[o]= MN ? f3min(a,b,c) : f3max(a,b,c);
  }
}
template<bool MN>
__device__ __forceinline__ void red_y(const float* __restrict__ s, float* __restrict__ d,
                                      int nz,int ny,int xlo,int xhi,int tid){
  const int nx=xhi-xlo+1, tot=nz*ny*nx;
  for (int i=tid;i<tot;i+=SBLK){
    int x=xlo+i%nx; int r=i/nx; int y=r%ny; int z=r/ny;
    int so=(z*(ny+2)+y)*PIT+x;
    float a=s[so],b=s[so+PIT],c=s[so+2*PIT];
    d[(z*ny+y)*PIT+x]= MN ? f3min(a,b,c) : f3max(a,b,c);
  }
}
template<bool MN>
__device__ __forceinline__ void red_x(const float* __restrict__ s, float* __restrict__ d,
                                      int nz,int ny,int xlo,int xhi,int tid){
  const int nx=xhi-xlo+1, tot=nz*ny*nx;
  for (int i=tid;i<tot;i+=SBLK){
    int x=xlo+i%nx; int r=i/nx; int y=r%ny; int z=r/ny;
    int o=(z*ny+y)*PIT+x;
    float a=s[o-1],b=s[o],c=s[o+1];
    d[o]= MN ? f3min(a,b,c) : f3max(a,b,c);
  }
}

// Re-pad intermediates outside the volume on boundary tiles:
// erode pads +inf, dilate pads -inf.
__device__ __forceinline__ void mask_oob(float* buf,int nzy,int xlo,int xhi,
                                         int gz0,int gy0,int gx0,float pad,int tid){
  const int nx=xhi-xlo+1, tot=nzy*nzy*nx;
  for (int i=tid;i<tot;i+=SBLK){
    int x=i%nx; int r=i/nx; int y=r%nzy; int z=r/nzy;
    if ((unsigned)(gz0+z)>=192u || (unsigned)(gy0+y)>=192u || (unsigned)(gx0+x)>=192u)
      buf[(z*nzy+y)*PIT+xlo+x]=pad;
  }
}

// Guarded (boundary-tile) loader with +inf padding.
__device__ __forceinline__ void load_tile_guarded(float* IN, const float* __restrict__ vbase,
                                                  int z0,int y0,int x0,int nzy,int oh,int tid){
  const int tot=nzy*nzy*PIT;
  for (int i=tid;i<tot;i+=SBLK){
    int x=i%PIT; int r=i/PIT; int zz=r/nzy, yy=r%nzy;
    int gz=z0-oh+zz, gy=y0-oh+yy, gx=x0-4+x;
    float v=PADPOS;
    if ((unsigned)gz<192u && (unsigned)gy<192u && (unsigned)gx<192u)
      v = vbase[(gz*192+gy)*192+gx];
    IN[r*PIT+x]=v;
  }
}

// Interior loader, init kernel: async DMA to LDS (ASYNCcnt), B128 chunks.
__device__ __forceinline__ void load_tile_async(float* IN, const float* __restrict__ vbase,
                                                int z0,int y0,int x0,int nzy,int oh,int tid){
  const int tot = nzy*nzy*(PIT/4);              // PIT/4 B128 chunks per row
  for (int c=tid;c<tot;c+=SBLK){
    int row=c/(PIT/4), xi=c%(PIT/4);
    int zz=row/nzy, yy=row%nzy;
    unsigned goff  = (unsigned)(((((z0-oh+zz)*192)+(y0-oh+yy))*192 + (x0-4+xi*4))*4);
    unsigned laddr = (unsigned)(uintptr_t)(IN + (row*PIT + xi*4));
    asm volatile("global_load_async_to_lds_b128 %0, %1, %2"
                 :: "v"(laddr), "v"(goff), "s"(vbase) : "memory");
  }
  asm volatile("s_wait_asynccnt 0" ::: "memory");
}

// Interior loader, iterate kernel: one Tensor Data Mover descriptor moves the
// whole PIT x nzy x nzy f32 tile (global strides 192/192^2) to LDS linearly.
// D# layout per CDNA5 ISA ch.8 (group0..group3). Issued by wave 0 only
// (EXEC-independent, TENSORcnt-tracked); tensor dims maxed so no OOB clipping
// (interior tiles are in-bounds by construction).
__device__ __forceinline__ void load_tile_tdm(float* IN, const float* __restrict__ gtile,
                                              int nzy,int tid){
  if (tid < 32) {
    unsigned lds_addr = (unsigned)(uintptr_t)IN;
    unsigned long long ga = (unsigned long long)(uintptr_t)gtile;
    u32x4 g0;
    g0.x = 1u;                                   // count=1, is_restore=0, gather=0
    g0.y = lds_addr;                             // lds_addr      [63:32]
    g0.z = (unsigned)(ga & 0xffffffffu);         // global_addr   [95:64]
    g0.w = (unsigned)((ga >> 32) & 0x01ffffffu)  // global_addr   [120:96]
         | (2u << 30);                           // type=2 (image) [127:126]
    u32x8 g1;
    g1.s0 = (2u << 16);                          // wg_mask=0, data_size=4B, no flags
    g1.s1 = 0xffffu << 16;                       // tensor_dim0[15:0]  (0x7fffffff)
    g1.s2 = 0x7fffu | (0xffffu << 16);           // tensor_dim0[31:16] | tensor_dim1[15:0]
    g1.s3 = 0x7fffu | ((unsigned)PIT << 16);     // tensor_dim1[31:16] | tile_dim0=PIT
    g1.s4 = (unsigned)nzy | ((unsigned)nzy<<16); // tile_dim1 | tile_dim2
    g1.s5 = 192u;                                // tensor_dim0_stride[31:0]
    g1.s6 = (36864u & 0xffffu) << 16;            // stride0[47:32]=0 | stride1[15:0]
    g1.s7 = 36864u >> 16;                        // tensor_dim1_stride[47:16]
    u32x4 g2; g2.x = 0x7fffffffu; g2.y = 0u; g2.z = 0u; g2.w = 0u; // tensor_dim2 big
    u32x4 g3; g3.x = 0u; g3.y = 0u; g3.z = 0u; g3.w = 0u;
    asm volatile("tensor_load_to_lds %0, %1, %2, %3"
                 :: "s"(g0), "s"(g1), "s"(g2), "s"(g3) : "memory");
    __builtin_amdgcn_s_wait_tensorcnt(0);
  }
}

// img[0..2N) = sigmoid(l1-l0) (softmax ch1), img[2N..4N) = (targets==1)
__global__ void __launch_bounds__(NTHR)
prologue_kernel(const float* __restrict__ logits, const int* __restrict__ targets,
                float* __restrict__ img){
  int q = blockIdx.x*NTHR + threadIdx.x;          // quad index over 4N/4
  if (q >= NVOX) return;                          // 4N/4 == NVOX
  int i = q*4;
  float4 v;
  if (i < 2*NVOX){
    int b=i/NVOX, r=i-b*NVOX;
    const float4 l0 = *(const float4*)&logits[(b*2+0)*NVOX+r];
    const float4 l1 = *(const float4*)&logits[(b*2+1)*NVOX+r];
    v.x = 1.0f/(1.0f+expf(l0.x-l1.x));
    v.y = 1.0f/(1.0f+expf(l0.y-l1.y));
    v.z = 1.0f/(1.0f+expf(l0.z-l1.z));
    v.w = 1.0f/(1.0f+expf(l0.w-l1.w));
  } else {
    const int4 t = *(const int4*)&targets[i-2*NVOX];
    v.x=(t.x==1)?1.0f:0.0f; v.y=(t.y==1)?1.0f:0.0f;
    v.z=(t.z==1)?1.0f:0.0f; v.w=(t.w==1)?1.0f:0.0f;
  }
  *(float4*)&img[i] = v;
}

// skel = relu(img - dilate(erode(img)))   (halo-2 fused open; async-LDS path)
// LDS (dynamic): R1=IN(12,12)[keep] R2=A->F R3=B->G R4=E->O
__global__ void __launch_bounds__(SBLK)
init_skel_kernel(const float* __restrict__ img, float* __restrict__ skel){
  extern __shared__ __align__(16) float lds[];
  float* R1=lds;                 // IN 12*12*40 = 5760 (kept until the end)
  float* R2=lds+5760;            // A 10*12*40 = 4800 -> F 8*10*40
  float* R3=lds+5760+4800;       // B 10*10*40 = 4000 -> G 8*8*40
  float* R4=lds+5760+4800+4000;  // E 10*10*40 = 4000 -> O 8*8*40
  const int tid=threadIdx.x;
  const int tz=blockIdx.z%24, vol=blockIdx.z/24;
  const int x0=blockIdx.x*TX, y0=blockIdx.y*TZY, z0=tz*TZY;
  const bool interior=(blockIdx.x>=1&&blockIdx.x<=4&&blockIdx.y>=1&&blockIdx.y<=22&&tz>=1&&tz<=22);
  const float* vimg = img + (size_t)vol*NVOX;
  if (interior) load_tile_async(R1, vimg, z0,y0,x0, 12, 2, tid);
  else          load_tile_guarded(R1, vimg, z0,y0,x0, 12, 2, tid);
  __syncthreads();
  red_z<true >(R1,R2,10,12,2,37,tid); __syncthreads();   // min over z
  red_y<true >(R2,R3,10,10,2,37,tid); __syncthreads();   // min over y
  red_x<true >(R3,R4,10,10,3,36,tid); __syncthreads();   // E = erode(img)
  if(!interior){ mask_oob(R4,10,3,36, z0-1,y0-1,x0-1, PADNEG, tid); __syncthreads(); }
  red_z<false>(R4,R2, 8,10,3,36,tid); __syncthreads();
  red_y<false>(R2,R3, 8, 8,3,36,tid); __syncthreads();
  red_x<false>(R3,R4, 8, 8,4,35,tid); __syncthreads();   // O = dilate(E)
  float* skv = skel + (size_t)vol*NVOX;
  for (int i=tid;i<512;i+=SBLK){                         // 512 quads = 2048 voxels
    int x4=(i&7)*4, y=(i>>3)&7, z=i>>6;
    const float4 c = *(const float4*)&R1[((z+2)*12+(y+2))*PIT+(x4+4)];
    const float4 o = *(const float4*)&R4[(z*8+y)*PIT+(x4+4)];
    float4 d;
    d.x=fmaxf(c.x-o.x,0.f); d.y=fmaxf(c.y-o.y,0.f);
    d.z=fmaxf(c.z-o.z,0.f); d.w=fmaxf(c.w-o.w,0.f);
    *(float4*)&skv[((size_t)(z0+z)*192+(y0+y))*192+(x0+x4)] = d;
  }
}

// One skeletonize iteration, fully fused with halo 3 (TDM-staged tile):
//   imgnew = erode(img); O = dilate(erode(imgnew));
//   delta = relu(imgnew - O); skel += relu(delta - skel*delta)
// LDS (dynamic): R1=IN(14,14)->C->F R2=A->D->G R3=B->E2->O R4=E1 (=imgnew)
__global__ void __launch_bounds__(SBLK)
iterate_kernel(const float* __restrict__ imgin, float* __restrict__ imgout,
               float* __restrict__ skel){
  extern __shared__ __align__(16) float lds[];
  float* R1=lds;                 // IN 14*14*40 = 7840 -> C 10*12*40 -> F 8*10*40
  float* R2=lds+7840;            // A 12*14*40 = 6720 -> D 10*10*40 -> G 8*8*40
  float* R3=lds+7840+6720;       // B 12*12*40 = 5760 -> E2 10*10*40 -> O 8*8*40
  float* R4=lds+7840+6720+5760;  // E1 12*12*40 = 5760 (kept; it is imgnew)
  const int tid=threadIdx.x;
  const int tz=blockIdx.z%24, vol=blockIdx.z/24;
  const int x0=blockIdx.x*TX, y0=blockIdx.y*TZY, z0=tz*TZY;
  const bool interior=(blockIdx.x>=1&&blockIdx.x<=4&&blockIdx.y>=1&&blockIdx.y<=22&&tz>=1&&tz<=22);
  const float* vin = imgin + (size_t)vol*NVOX;
  if (interior) load_tile_tdm(R1, vin + (((z0-3)*192+(y0-3))*192 + (x0-4)), 14, tid);
  else          load_tile_guarded(R1, vin, z0,y0,x0, 14, 3, tid);
  __syncthreads();
  red_z<true >(R1,R2,12,14,1,38,tid); __syncthreads();
  red_y<true >(R2,R3,12,12,1,38,tid); __syncthreads();
  red_x<true >(R3,R4,12,12,2,37,tid); __syncthreads();   // E1 = erode(img) = imgnew
  if(!interior){ mask_oob(R4,12,2,37, z0-2,y0-2,x0-2, PADPOS, tid); __syncthreads(); }
  red_z<true >(R4,R1,10,12,2,37,tid); __syncthreads();
  red_y<true >(R1,R2,10,10,2,37,tid); __syncthreads();
  red_x<true >(R2,R3,10,10,3,36,tid); __syncthreads();   // E2 = erode(E1)
  if(!interior){ mask_oob(R3,10,3,36, z0-1,y0-1,x0-1, PADNEG, tid); __syncthreads(); }
  red_z<false>(R3,R1, 8,10,3,36,tid); __syncthreads();
  red_y<false>(R1,R2, 8, 8,3,36,tid); __syncthreads();
  red_x<false>(R2,R3, 8, 8,4,35,tid); __syncthreads();   // O = dilate(E2)
  float* sv = skel   + (size_t)vol*NVOX;
  float* ov = imgout + (size_t)vol*NVOX;
  for (int i=tid;i<512;i+=SBLK){                         // 512 quads = 2048 voxels
    int x4=(i&7)*4, y=(i>>3)&7, z=i>>6;
    const float4 e1 = *(const float4*)&R4[((z+2)*12+(y+2))*PIT+(x4+4)];
    const float4 o  = *(const float4*)&R3[(z*8+y)*PIT+(x4+4)];
    size_t g = ((size_t)(z0+z)*192+(y0+y))*192+(x0+x4);
    float4 s = *(const float4*)&sv[g];
    float4 d, u;
    d.x=fmaxf(e1.x-o.x,0.f); d.y=fmaxf(e1.y-o.y,0.f);
    d.z=fmaxf(e1.z-o.z,0.f); d.w=fmaxf(e1.w-o.w,0.f);
    u.x=s.x+fmaxf(d.x-s.x*d.x,0.f); u.y=s.y+fmaxf(d.y-s.y*d.y,0.f);
    u.z=s.z+fmaxf(d.z-s.z*d.z,0.f); u.w=s.w+fmaxf(d.w-s.w*d.w,0.f);
    *(float4*)&sv[g] = u;
    *(float4*)&ov[g] = e1;
  }
}

// Per-block partial sums (deterministic; pred/gt recomputed from the inputs).
__global__ void __launch_bounds__(NTHR)
reduce_kernel(const float* __restrict__ skel, const float* __restrict__ logits,
              const int* __restrict__ targets, float* __restrict__ partials){
  float s0=0.f,s1=0.f,s2=0.f,s3=0.f;
  const int totq = (2*NVOX)/4;
  for (int q=blockIdx.x*NTHR+threadIdx.x; q<totq; q+=NRBLK*NTHR){
    int i=q*4;
    int b=i/NVOX, r=i-b*NVOX;
    const float4 sp = *(const float4*)&skel[i];
    const float4 sg = *(const float4*)&skel[2*NVOX+i];
    const float4 l0 = *(const float4*)&logits[(b*2+0)*NVOX+r];
    const float4 l1 = *(const float4*)&logits[(b*2+1)*NVOX+r];
    const int4   t  = *(const int4*)&targets[i];
    float4 pr;
    pr.x=1.0f/(1.0f+expf(l0.x-l1.x)); pr.y=1.0f/(1.0f+expf(l0.y-l1.y));
    pr.z=1.0f/(1.0f+expf(l0.z-l1.z)); pr.w=1.0f/(1.0f+expf(l0.w-l1.w));
    s0 += sp.x+sp.y+sp.z+sp.w;
    s1 += sp.x*((t.x==1)?1.f:0.f)+sp.y*((t.y==1)?1.f:0.f)
        + sp.z*((t.z==1)?1.f:0.f)+sp.w*((t.w==1)?1.f:0.f);
    s2 += sg.x+sg.y+sg.z+sg.w;
    s3 += sg.x*pr.x+sg.y*pr.y+sg.z*pr.z+sg.w*pr.w;
  }
  __shared__ float red[NTHR];
  float v[4]={s0,s1,s2,s3};
  for (int k=0;k<4;k++){
    red[threadIdx.x]=v[k]; __syncthreads();
    for (int s=NTHR/2;s>0;s>>=1){ if(threadIdx.x<s) red[threadIdx.x]+=red[threadIdx.x+s]; __syncthreads(); }
    if (threadIdx.x==0) partials[k*NRBLK+blockIdx.x]=red[0];
    __syncthreads();
  }
}

__global__ void __launch_bounds__(NTHR)
finalize_kernel(const float* __restrict__ partials, float* __restrict__ out){
  __shared__ float red[NTHR];
  float sums[4];
  for (int k=0;k<4;k++){
    float a=0.f;
    for (int i=threadIdx.x;i<NRBLK;i+=NTHR) a+=partials[k*NRBLK+i];
    red[threadIdx.x]=a; __syncthreads();
    for (int s=NTHR/2;s>0;s>>=1){ if(threadIdx.x<s) red[threadIdx.x]+=red[threadIdx.x+s]; __syncthreads(); }
    sums[k]=red[0]; __syncthreads();
  }
  if (threadIdx.x==0){
    float tprec=(sums[1]+1.0f)/(sums[0]+1.0f);
    float tsens=(sums[3]+1.0f)/(sums[2]+1.0f);
    float cl=2.0f*tprec*tsens/(tprec+tsens+1e-7f);
    out[0]=1.0f-cl;
  }
}

extern "C" void kernel_launch(void* const* d_in, const int* in_sizes, int n_in,
                              void* d_out, int out_size, void* d_ws, size_t ws_size,
                              hipStream_t stream){
  const float* logits  = (const float*)d_in[0];   // (2,2,192,192,192) f32
  const int*   targets = (const int*)  d_in[1];   // (2,192,192,192)   i32
  float* out = (float*)d_out;                     // scalar loss
  float* ws  = (float*)d_ws;
  float* imgA     = ws;                           // 4N
  float* imgB     = ws + (size_t)4*NVOX;          // 4N
  float* skel     = ws + (size_t)8*NVOX;          // 4N
  float* partials = ws + (size_t)12*NVOX;         // 4*NRBLK

  prologue_kernel<<<(NVOX+NTHR-1)/NTHR, NTHR, 0, stream>>>(logits, targets, imgA);

  dim3 grid(192/TX, 192/TZY, (192/TZY)*4);  // (6, 24, 24*4 volumes)
  const size_t initLds = (size_t)(5760+4800+4000+4000)*sizeof(float);   //  74.2 KB
  const size_t iterLds = (size_t)(7840+6720+5760+5760)*sizeof(float);   // 104.3 KB

  init_skel_kernel<<<grid, SBLK, initLds, stream>>>(imgA, skel);

  float* cur=imgA; float* nxt=imgB;
  for (int it=0; it<10; ++it){
    iterate_kernel<<<grid, SBLK, iterLds, stream>>>(cur, nxt, skel);
    float* t=cur; cur=nxt; nxt=t;
  }

  reduce_kernel<<<NRBLK, NTHR, 0, stream>>>(skel, logits, targets, partials);
  finalize_kernel<<<1, NTHR, 0, stream>>>(partials, out);
}